// DiagonalSSM_35003983462988
// MI455X (gfx1250) — compile-verified
//
#include <hip/hip_runtime.h>

typedef __attribute__((ext_vector_type(2))) float v2f;
typedef __attribute__((ext_vector_type(8))) float v8f;

#define B_ 256
#define T_ 4096
#define N_ 64
#define L_ 64
#define C_ 64   // T_/L_

// workspace layout (float offsets)
#define OFF_MCONVT 0                      // [s][t] = k[t-s], 64x64
#define OFF_GRT    4096                   // [n][t] = Re(conj(c_n)*lam_n^{t+1})
#define OFF_GINT   8192                   // [n][t] = -Im(conj(c_n)*lam_n^{t+1})
#define OFF_AR     12288                  // [s][n] = Re(b_n*lam_n^{63-s})
#define OFF_AI     16384                  // [s][n] = Im(b_n*lam_n^{63-s})
#define OFF_LAMR   20480                  // lam^64 real
#define OFF_LAMI   20544                  // lam^64 imag
#define OFF_WXR    20608                  // [b][c][n]  W -> chunk-initial state (real)
#define OFF_WXI    (20608 + B_*C_*N_)     // imag plane

// ---------------------------------------------------------------- kernel 0
// Precompute lambda, conv kernel matrix, input/readout matrices. 64 threads.
__global__ void ssm_prep(const float* __restrict__ rho, const float* __restrict__ theta,
                         const float* __restrict__ br_, const float* __restrict__ bi_,
                         const float* __restrict__ cr_, const float* __restrict__ ci_,
                         float* __restrict__ ws) {
  __shared__ float sh[64 * 64];
  __shared__ float kv[64];
  const int n = threadIdx.x;              // 0..63 (state index / later column index)
  const float rh = rho[n];
  const float sp = fmaxf(rh, 0.0f) + log1pf(expf(-fabsf(rh)));   // softplus
  const float r  = expf(-sp);
  const float th = theta[n];
  const float lre = r * cosf(th);
  const float lim = r * sinf(th);
  const float brn = br_[n], bin = bi_[n];
  const float crn = cr_[n], cin = ci_[n];
  // cb = conj(c)*b
  const float cbr = crn * brn + cin * bin;
  const float cbi = crn * bin - cin * brn;

  // powers p = lam^e ; fill per-n kernel contributions and A matrices
  float pr = 1.0f, pi = 0.0f;
  for (int e = 0; e < 64; ++e) {
    sh[e * 64 + n] = cbr * pr - cbi * pi;         // Re(cb * lam^e)
    const int s = 63 - e;
    ws[OFF_AR + s * 64 + n] = brn * pr - bin * pi;
    ws[OFF_AI + s * 64 + n] = brn * pi + bin * pr;
    const float npr = lre * pr - lim * pi;
    const float npi = lre * pi + lim * pr;
    pr = npr; pi = npi;
  }
  ws[OFF_LAMR + n] = pr;                          // lam^64
  ws[OFF_LAMI + n] = pi;

  // G rows: q = lam^{t+1}
  float qr = lre, qi = lim;
  for (int t = 0; t < 64; ++t) {
    ws[OFF_GRT  + n * 64 + t] = crn * qr + cin * qi;   //  Re(conj(c)*q)
    ws[OFF_GINT + n * 64 + t] = cin * qr - crn * qi;   // -Im(conj(c)*q)
    const float nqr = lre * qr - lim * qi;
    const float nqi = lre * qi + lim * qr;
    qr = nqr; qi = nqi;
  }
  __syncthreads();

  // k[d] = sum_n Re(cb_n * lam_n^d) ; thread n reduces row d=n
  float acc = 0.0f;
  for (int j = 0; j < 64; ++j) acc += sh[n * 64 + j];
  kv[n] = acc;
  __syncthreads();

  // Toeplitz conv matrix, transposed for GEMM: MconvT[s][t] = k[t-s] (t>=s)
  const int t = n;
  for (int s = 0; s < 64; ++s)
    ws[OFF_MCONVT + s * 64 + t] = (t >= s) ? kv[t - s] : 0.0f;
}

// ---------------------------------------------------------------- kernel 1
// W[b,c,n] = sum_s u[b, c*64+s] * A[s][n]   (real & imag via blockIdx.z)
__global__ __launch_bounds__(128)
void ssm_chunk_w(const float* __restrict__ u, float* __restrict__ ws) {
  __shared__ float Ulds[16 * 64];
  const int m0 = blockIdx.x * 16;         // batch-row tile
  const int c  = blockIdx.y;              // chunk
  const int z  = blockIdx.z;              // 0=real 1=imag
  const float* __restrict__ Amat = ws + (z ? OFF_AI : OFF_AR);
  float* __restrict__ out        = ws + (z ? OFF_WXI : OFF_WXR);
  const int tid = threadIdx.x;
  for (int idx = tid; idx < 16 * 64; idx += 128) {
    const int row = idx >> 6, col = idx & 63;
    Ulds[idx] = u[(m0 + row) * T_ + c * 64 + col];
  }
  __syncthreads();
  const int wave = tid >> 5;
  const int lane = tid & 31;
  const int half = lane >> 4;             // lanes 16-31 carry K+2,K+3
  const int lr   = lane & 15;
  const int n0   = wave * 16;
  v8f acc = {};
  for (int k0 = 0; k0 < 64; k0 += 4) {
    const int ka = k0 + 2 * half;
    v2f a, b;
    a.x = Ulds[lr * 64 + ka];
    a.y = Ulds[lr * 64 + ka + 1];
    b.x = Amat[ka * 64 + n0 + lr];
    b.y = Amat[(ka + 1) * 64 + n0 + lr];
    acc = __builtin_amdgcn_wmma_f32_16x16x4_f32(false, a, false, b, (short)0, acc, false, false);
  }
  for (int j = 0; j < 8; ++j) {
    const int row = m0 + j + half * 8;
    out[(row * C_ + c) * 64 + n0 + lr] = acc[j];
  }
}

// ---------------------------------------------------------------- kernel 2
// Cross-chunk scan (64 steps). Overwrites W planes in place with the
// chunk-INITIAL state x0 for each chunk: x_{c+1} = lam^64 * x_c + W_c.
__global__ __launch_bounds__(256)
void ssm_scan(float* __restrict__ ws) {
  const int gid = blockIdx.x * 256 + threadIdx.x;  // b*64 + n
  const int b = gid >> 6;
  const int n = gid & 63;
  const float lamr = ws[OFF_LAMR + n];
  const float lami = ws[OFF_LAMI + n];
  float* __restrict__ Wr = ws + OFF_WXR;
  float* __restrict__ Wi = ws + OFF_WXI;
  float xr = 0.0f, xi = 0.0f;
  for (int c = 0; c < C_; ++c) {
    const int idx = (b * C_ + c) * 64 + n;
    const float wr = Wr[idx], wi = Wi[idx];
    Wr[idx] = xr; Wi[idx] = xi;                    // store chunk-initial state
    const float nxr = lamr * xr - lami * xi + wr;
    const float nxi = lamr * xi + lami * xr + wi;
    xr = nxr; xi = nxi;
  }
}

// ---------------------------------------------------------------- kernel 3
// y[b, c*64+t] = sum_s MconvT[s][t]*u[b,c*64+s]
//              + sum_n GrT[n][t]*x0r[b,c,n] + GiNegT[n][t]*x0i[b,c,n]
__global__ __launch_bounds__(128)
void ssm_out(const float* __restrict__ u, const float* __restrict__ ws,
             float* __restrict__ y) {
  __shared__ float Ulds[16 * 64];
  __shared__ float Xr[16 * 64];
  __shared__ float Xi[16 * 64];
  const int m0 = blockIdx.x * 16;
  const int c  = blockIdx.y;
  const int tid = threadIdx.x;
  const float* __restrict__ WXr = ws + OFF_WXR;
  const float* __restrict__ WXi = ws + OFF_WXI;
  for (int idx = tid; idx < 16 * 64; idx += 128) {
    const int row = idx >> 6, col = idx & 63;
    Ulds[idx] = u[(m0 + row) * T_ + c * 64 + col];
    Xr[idx]   = WXr[((m0 + row) * C_ + c) * 64 + col];
    Xi[idx]   = WXi[((m0 + row) * C_ + c) * 64 + col];
  }
  __syncthreads();
  const int wave = tid >> 5;
  const int lane = tid & 31;
  const int half = lane >> 4;
  const int lr   = lane & 15;
  const int t0   = wave * 16;
  const float* __restrict__ Mc = ws + OFF_MCONVT;
  const float* __restrict__ Gr = ws + OFF_GRT;
  const float* __restrict__ Gi = ws + OFF_GINT;
  v8f acc = {};
  for (int k0 = 0; k0 < 64; k0 += 4) {
    const int ka = k0 + 2 * half;
    v2f a, b;
    // segment 1: causal Toeplitz convolution of u with k
    a.x = Ulds[lr * 64 + ka];      a.y = Ulds[lr * 64 + ka + 1];
    b.x = Mc[ka * 64 + t0 + lr];   b.y = Mc[(ka + 1) * 64 + t0 + lr];
    acc = __builtin_amdgcn_wmma_f32_16x16x4_f32(false, a, false, b, (short)0, acc, false, false);
    // segment 2: Re-part of chunk-initial-state correction
    a.x = Xr[lr * 64 + ka];        a.y = Xr[lr * 64 + ka + 1];
    b.x = Gr[ka * 64 + t0 + lr];   b.y = Gr[(ka + 1) * 64 + t0 + lr];
    acc = __builtin_amdgcn_wmma_f32_16x16x4_f32(false, a, false, b, (short)0, acc, false, false);
    // segment 3: Im-part (G already negated)
    a.x = Xi[lr * 64 + ka];        a.y = Xi[lr * 64 + ka + 1];
    b.x = Gi[ka * 64 + t0 + lr];   b.y = Gi[(ka + 1) * 64 + t0 + lr];
    acc = __builtin_amdgcn_wmma_f32_16x16x4_f32(false, a, false, b, (short)0, acc, false, false);
  }
  for (int j = 0; j < 8; ++j) {
    const int row = m0 + j + half * 8;
    y[row * T_ + c * 64 + t0 + lr] = acc[j];
  }
}

// ---------------------------------------------------------------- launch
extern "C" void kernel_launch(void* const* d_in, const int* in_sizes, int n_in,
                              void* d_out, int out_size, void* d_ws, size_t ws_size,
                              hipStream_t stream) {
  (void)in_sizes; (void)n_in; (void)out_size; (void)ws_size;
  const float* u     = (const float*)d_in[0];
  const float* rho   = (const float*)d_in[1];
  const float* theta = (const float*)d_in[2];
  const float* br    = (const float*)d_in[3];
  const float* bi    = (const float*)d_in[4];
  const float* cr    = (const float*)d_in[5];
  const float* ci    = (const float*)d_in[6];
  float* y  = (float*)d_out;
  float* ws = (float*)d_ws;

  ssm_prep<<<1, 64, 0, stream>>>(rho, theta, br, bi, cr, ci, ws);
  ssm_chunk_w<<<dim3(B_ / 16, C_, 2), 128, 0, stream>>>(u, ws);
  ssm_scan<<<dim3((B_ * N_) / 256), 256, 0, stream>>>(ws);
  ssm_out<<<dim3(B_ / 16, C_), 128, 0, stream>>>(u, ws, y);
}